// Test_23493471109622
// MI455X (gfx1250) — compile-verified
//
#include <hip/hip_runtime.h>

typedef float v2f __attribute__((ext_vector_type(2)));
typedef float v8f __attribute__((ext_vector_type(8)));

// ---------------------------------------------------------------------------
// Utility kernels
// ---------------------------------------------------------------------------
__global__ __launch_bounds__(256) void zero_f32(float* __restrict__ p, int count) {
    int i4 = blockIdx.x * blockDim.x + threadIdx.x;
    const int stride = gridDim.x * blockDim.x;
    const int n4 = count >> 2;
    float4* p4 = (float4*)p;
    for (int i = i4; i < n4; i += stride) p4[i] = make_float4(0.f, 0.f, 0.f, 0.f);
    // tail
    for (int i = (n4 << 2) + i4; i < count; i += stride) p[i] = 0.0f;
}

__global__ __launch_bounds__(256) void degree_kernel(const int* __restrict__ src,
                                                     const int* __restrict__ dst,
                                                     float* __restrict__ out_deg,
                                                     float* __restrict__ in_deg,
                                                     int nE) {
    int e = blockIdx.x * blockDim.x + threadIdx.x;
    if (e >= nE) return;
    atomicAdd(&out_deg[src[e]], 1.0f);   // non-returning global_atomic_add_f32
    atomicAdd(&in_deg[dst[e]], 1.0f);
}

__global__ __launch_bounds__(256) void norm_kernel(float* __restrict__ deg, int count) {
    int i = blockIdx.x * blockDim.x + threadIdx.x;
    if (i >= count) return;
    deg[i] = rsqrtf(fmaxf(deg[i], 1.0f));
}

// ---------------------------------------------------------------------------
// WMMA node transforms (fp32, V_WMMA_F32_16X16X4_F32, exact precision).
// One wave32 per 16-node tile; 8 waves / 256-thread block.
// A layout: lane L holds row M=L&15, K = kbase + 2*(L>>4) + {0,1}
// B layout: lane L holds col N=L&15, same K pair (staged in LDS pair layout)
// D layout: lane L col N=L&15, VGPR v holds row M = v + 8*(L>>4)
// All operand loads are branchless (clamped address + cndmask zeroing) so the
// wave reaches every WMMA with EXEC all-1s and no saveexec churn.
// ---------------------------------------------------------------------------

// h1[n,0:8] = (feat[n,0:10] @ W1[10,8]) * out_norm[n]
__global__ __launch_bounds__(256) void node_transform1(const float* __restrict__ feat,
                                                       const float* __restrict__ W1,
                                                       const float* __restrict__ out_norm,
                                                       float* __restrict__ h1,
                                                       int nNodes) {
    __shared__ float wpair[6 * 16 * 2];          // [k/2][n][2], zero padded K->12, N->16
    for (int idx = threadIdx.x; idx < 6 * 16 * 2; idx += 256) {
        const int j = idx & 1, mm = (idx >> 1) & 15, kk = idx >> 5;
        const int k = kk * 2 + j;
        wpair[idx] = (k < 10 && mm < 8) ? W1[k * 8 + mm] : 0.0f;
    }
    __syncthreads();

    const int lane = threadIdx.x & 31;
    const int wid  = threadIdx.x >> 5;
    const int row0 = (blockIdx.x * 8 + wid) * 16;
    const int m    = lane & 15;                  // A row in tile / B column
    const int hi   = lane >> 4;                  // K-half select
    const int node = row0 + m;
    const int nodeC = min(node, nNodes - 1);
    const bool nodeOk = node < nNodes;
    const v2f zero2 = {0.0f, 0.0f};

    v8f acc = {};
#pragma unroll
    for (int s = 0; s < 3; ++s) {                // K = 10 padded to 12
        const int k0 = s * 4 + hi * 2;
        int off = nodeC * 10 + k0;               // 8B aligned (k0 even, row stride even)
        off = min(off, nNodes * 10 - 2);         // clamp (k0==10 tail)
        v2f a = *reinterpret_cast<const v2f*>(feat + off);
        a = (nodeOk && k0 < 10) ? a : zero2;     // branchless zeroing
        const int kk = s * 2 + hi;
        const v2f b = *reinterpret_cast<const v2f*>(&wpair[(kk * 16 + m) * 2]);
        acc = __builtin_amdgcn_wmma_f32_16x16x4_f32(false, a, false, b,
                                                    (short)0, acc, false, false);
    }

    const bool fullTile = (row0 + 16) <= nNodes; // wave-uniform
    if (m < 8) {
        const int rb = row0 + hi * 8;
        if (fullTile) {
#pragma unroll
            for (int v = 0; v < 8; ++v)
                h1[(rb + v) * 8 + m] = acc[v] * out_norm[rb + v];
        } else {
#pragma unroll
            for (int v = 0; v < 8; ++v)
                if (rb + v < nNodes) h1[(rb + v) * 8 + m] = acc[v] * out_norm[rb + v];
        }
    }
}

// h2[n,0:4] = ((agg1[n,0:8]*in_norm[n] + b1) @ W2[8,4]) * out_norm[n]
__global__ __launch_bounds__(256) void node_transform2(const float* __restrict__ agg1,
                                                       const float* __restrict__ W2,
                                                       const float* __restrict__ b1,
                                                       const float* __restrict__ in_norm,
                                                       const float* __restrict__ out_norm,
                                                       float* __restrict__ h2,
                                                       int nNodes) {
    __shared__ float wpair[4 * 16 * 2];          // [k/2][n][2], N padded 4->16
    __shared__ float bpair[4 * 2];               // b1 pairs
    for (int idx = threadIdx.x; idx < 4 * 16 * 2; idx += 256) {
        const int j = idx & 1, mm = (idx >> 1) & 15, kk = idx >> 5;
        const int k = kk * 2 + j;
        wpair[idx] = (mm < 4) ? W2[k * 4 + mm] : 0.0f;
    }
    if (threadIdx.x < 8) bpair[threadIdx.x] = b1[threadIdx.x];
    __syncthreads();

    const int lane = threadIdx.x & 31;
    const int wid  = threadIdx.x >> 5;
    const int row0 = (blockIdx.x * 8 + wid) * 16;
    const int m    = lane & 15;
    const int hi   = lane >> 4;
    const int node = row0 + m;
    const int nodeC = min(node, nNodes - 1);
    const bool nodeOk = node < nNodes;
    const float inn = in_norm[nodeC];
    const v2f zero2 = {0.0f, 0.0f};

    v8f acc = {};
#pragma unroll
    for (int s = 0; s < 2; ++s) {                // K = 8
        const int k0 = s * 4 + hi * 2;
        const int kk = s * 2 + hi;
        v2f g  = *reinterpret_cast<const v2f*>(agg1 + nodeC * 8 + k0);  // aligned b64
        const v2f bb = *reinterpret_cast<const v2f*>(&bpair[kk * 2]);
        v2f a;
        a.x = g.x * inn + bb.x;
        a.y = g.y * inn + bb.y;
        a = nodeOk ? a : zero2;
        const v2f b = *reinterpret_cast<const v2f*>(&wpair[(kk * 16 + m) * 2]);
        acc = __builtin_amdgcn_wmma_f32_16x16x4_f32(false, a, false, b,
                                                    (short)0, acc, false, false);
    }

    const bool fullTile = (row0 + 16) <= nNodes;
    if (m < 4) {
        const int rb = row0 + hi * 8;
        if (fullTile) {
#pragma unroll
            for (int v = 0; v < 8; ++v)
                h2[(rb + v) * 4 + m] = acc[v] * out_norm[rb + v];
        } else {
#pragma unroll
            for (int v = 0; v < 8; ++v)
                if (rb + v < nNodes) h2[(rb + v) * 4 + m] = acc[v] * out_norm[rb + v];
        }
    }
}

// out[n,0:2] = (agg2[n,0:4]*in_norm[n] + b2) @ Wl[4,2] + bl
__global__ __launch_bounds__(256) void node_final(const float* __restrict__ agg2,
                                                  const float* __restrict__ Wl,
                                                  const float* __restrict__ b2,
                                                  const float* __restrict__ bl,
                                                  const float* __restrict__ in_norm,
                                                  float* __restrict__ out,
                                                  int nNodes) {
    __shared__ float wpair[2 * 16 * 2];          // [k/2][n][2], N padded 2->16
    __shared__ float bpair[2 * 2];               // b2 pairs
    __shared__ float blpad[16];                  // bl padded to 16 cols
    for (int idx = threadIdx.x; idx < 2 * 16 * 2; idx += 256) {
        const int j = idx & 1, mm = (idx >> 1) & 15, kk = idx >> 5;
        const int k = kk * 2 + j;
        wpair[idx] = (mm < 2) ? Wl[k * 2 + mm] : 0.0f;
    }
    if (threadIdx.x < 4)  bpair[threadIdx.x] = b2[threadIdx.x];
    if (threadIdx.x < 16) blpad[threadIdx.x] = (threadIdx.x < 2) ? bl[threadIdx.x] : 0.0f;
    __syncthreads();

    const int lane = threadIdx.x & 31;
    const int wid  = threadIdx.x >> 5;
    const int row0 = (blockIdx.x * 8 + wid) * 16;
    const int m    = lane & 15;
    const int hi   = lane >> 4;
    const int node = row0 + m;
    const int nodeC = min(node, nNodes - 1);
    const bool nodeOk = node < nNodes;
    const float inn = in_norm[nodeC];
    const v2f zero2 = {0.0f, 0.0f};

    const int k0 = hi * 2;                       // K = 4, single WMMA
    const int kk = hi;
    v2f g = *reinterpret_cast<const v2f*>(agg2 + nodeC * 4 + k0);
    const v2f bb = *reinterpret_cast<const v2f*>(&bpair[kk * 2]);
    v2f a;
    a.x = g.x * inn + bb.x;
    a.y = g.y * inn + bb.y;
    a = nodeOk ? a : zero2;
    const v2f b = *reinterpret_cast<const v2f*>(&wpair[(kk * 16 + m) * 2]);
    v8f acc = {};
    acc = __builtin_amdgcn_wmma_f32_16x16x4_f32(false, a, false, b,
                                                (short)0, acc, false, false);

    const bool fullTile = (row0 + 16) <= nNodes;
    if (m < 2) {
        const float bln = blpad[m];
        const int rb = row0 + hi * 8;
        if (fullTile) {
#pragma unroll
            for (int v = 0; v < 8; ++v)
                out[(rb + v) * 2 + m] = acc[v] + bln;
        } else {
#pragma unroll
            for (int v = 0; v < 8; ++v)
                if (rb + v < nNodes) out[(rb + v) * 2 + m] = acc[v] + bln;
        }
    }
}

// ---------------------------------------------------------------------------
// Edge scatter-add kernels (L2-resident: h/agg tables are 8-16MB << 192MB L2)
// ---------------------------------------------------------------------------
__global__ __launch_bounds__(256) void edge_prop8(const int* __restrict__ src,
                                                  const int* __restrict__ dst,
                                                  const float* __restrict__ h,
                                                  float* __restrict__ agg,
                                                  int nE) {
    int e = blockIdx.x * blockDim.x + threadIdx.x;
    if (e >= nE) return;
    const int s = src[e], d = dst[e];
    const float4 v0 = *(const float4*)(h + (size_t)s * 8);
    const float4 v1 = *(const float4*)(h + (size_t)s * 8 + 4);
    float* a = agg + (size_t)d * 8;
    atomicAdd(a + 0, v0.x); atomicAdd(a + 1, v0.y);
    atomicAdd(a + 2, v0.z); atomicAdd(a + 3, v0.w);
    atomicAdd(a + 4, v1.x); atomicAdd(a + 5, v1.y);
    atomicAdd(a + 6, v1.z); atomicAdd(a + 7, v1.w);
}

__global__ __launch_bounds__(256) void edge_prop4(const int* __restrict__ src,
                                                  const int* __restrict__ dst,
                                                  const float* __restrict__ h,
                                                  float* __restrict__ agg,
                                                  int nE) {
    int e = blockIdx.x * blockDim.x + threadIdx.x;
    if (e >= nE) return;
    const int s = src[e], d = dst[e];
    const float4 v0 = *(const float4*)(h + (size_t)s * 4);
    float* a = agg + (size_t)d * 4;
    atomicAdd(a + 0, v0.x); atomicAdd(a + 1, v0.y);
    atomicAdd(a + 2, v0.z); atomicAdd(a + 3, v0.w);
}

// ---------------------------------------------------------------------------
// Host-side launch
// ---------------------------------------------------------------------------
extern "C" void kernel_launch(void* const* d_in, const int* in_sizes, int n_in,
                              void* d_out, int out_size, void* d_ws, size_t ws_size,
                              hipStream_t stream) {
    const float* feat = (const float*)d_in[0];
    const int*   src  = (const int*)d_in[1];
    const int*   dst  = (const int*)d_in[2];
    const float* W1   = (const float*)d_in[3];
    const float* b1   = (const float*)d_in[4];
    const float* W2   = (const float*)d_in[5];
    const float* b2   = (const float*)d_in[6];
    const float* Wl   = (const float*)d_in[7];
    const float* bl   = (const float*)d_in[8];
    float* out = (float*)d_out;

    const int N = in_sizes[0] / 10;
    const int E = in_sizes[1];

    // Workspace layout (floats), total 18N = 36 MB:
    //   [0,N)      out_norm (deg accumulator, then rsqrt)
    //   [N,2N)     in_norm
    //   [2N,10N)   agg1 [N,8]
    //   [10N,18N)  h1 [N,8]; after edge1: low half reused as h2 [N,4],
    //              high half reused as agg2 [N,4]
    float* W = (float*)d_ws;
    float* out_norm = W;
    float* in_norm  = W + (size_t)N;
    float* agg1     = W + (size_t)2 * N;
    float* h1       = W + (size_t)10 * N;
    float* h2       = W + (size_t)10 * N;   // overlays dead h1
    float* agg2     = W + (size_t)14 * N;   // overlays dead h1 upper half

    const int T = 256;
    const int edgeBlocks = (E + T - 1) / T;
    const int tileBlocks = (N + 127) / 128;   // 8 waves * 16 nodes per block

    // 1) zero degree accumulators + agg1 (contiguous 10N floats)
    zero_f32<<<2048, T, 0, stream>>>(W, 10 * N);
    // 2) degrees via f32 atomics
    degree_kernel<<<edgeBlocks, T, 0, stream>>>(src, dst, out_norm, in_norm, E);
    // 3) deg -> rsqrt(max(deg,1)) for both arrays at once
    norm_kernel<<<(2 * N + T - 1) / T, T, 0, stream>>>(W, 2 * N);
    // 4) layer-1 node transform (WMMA f32) with src-norm
    node_transform1<<<tileBlocks, T, 0, stream>>>(feat, W1, out_norm, h1, N);
    // 5) layer-1 edge scatter-add
    edge_prop8<<<edgeBlocks, T, 0, stream>>>(src, dst, h1, agg1, E);
    // 6) fused dst-norm + bias + layer-2 transform (WMMA) + src-norm
    node_transform2<<<tileBlocks, T, 0, stream>>>(agg1, W2, b1, in_norm, out_norm, h2, N);
    // 7) zero agg2 (its region held stale h1 data)
    zero_f32<<<2048, T, 0, stream>>>(agg2, 4 * N);
    // 8) layer-2 edge scatter-add
    edge_prop4<<<edgeBlocks, T, 0, stream>>>(src, dst, h2, agg2, E);
    // 9) fused dst-norm + bias + final linear (WMMA) + bias
    node_final<<<tileBlocks, T, 0, stream>>>(agg2, Wl, b2, bl, in_norm, out, N);
}